// DWT1D_57140244906761
// MI455X (gfx1250) — compile-verified
//
#include <hip/hip_runtime.h>
#include <stdint.h>

typedef __attribute__((ext_vector_type(2))) float v2f;
typedef __attribute__((ext_vector_type(8))) float v8f;

#define DWT_N0   16384
#define DWT_N1   8197
#define DWT_N2   4104
#define DWT_N3   2057
#define DWT_HALF 10
#define DWT_L    12
#define HALO     16                      // zero-filled left halo (need 10)
#define BUFA_SZ  (HALO + DWT_N0 + 32)    // 16432 f32: x row (later reused for lo2)
#define BUFB_SZ  (HALO + 8240)           // 8256 f32: lo1 (8197) + zero tail to idx 8223+

// Compute one 16-output tile with 3 chained v_wmma_f32_16x16x4_f32:
// D(16x16) += A(16x4) x B(4x16); A rows 0/1 = h0/h1 chunks (rows 2..15 zero),
// B columns = 16 output positions => D row0 = lo, D row1 = hi.
__device__ __forceinline__ v8f dwt_tile_wmma(const float* __restrict__ p,
                                             const v2f* __restrict__ A)
{
    v8f c = {};
#pragma unroll
    for (int q = 0; q < 3; ++q) {
        v2f b;
        b.x = p[4 * q];        // B K-row kx   (kx = lane>>4)
        b.y = p[4 * q + 2];    // B K-row kx+2
        c = __builtin_amdgcn_wmma_f32_16x16x4_f32(
                false, A[q], false, b, (short)0, c, false, false);
    }
    return c;
}

// One analysis-filter-bank level.  src points at element 0 of a zero-halo'd
// LDS buffer so all tap loads are unconditional ds ops.  Full tiles store
// unconditionally (only the static lane<16 mask); the single partial tail
// tile (every level here has outLen % 16 != 0) is handled by one wave with
// a wave-uniform ownership test, so EXEC is all-ones around every WMMA.
__device__ __forceinline__ void dwt_level_wmma(
    const float* __restrict__ src, int outLen,
    float* __restrict__ dstLo, float* __restrict__ dstHi,
    const v2f* __restrict__ A, int lane, int wave)
{
    const int nFull = outLen >> 4;          // full 16-output tiles
    const int n  = lane & 15;               // B column (output position in tile)
    const int kx = lane >> 4;

    for (int t = wave; t < nFull; t += 8) {
        const int j0 = t << 4;
        const float* p = src + (2 * (j0 + n) + kx - DWT_HALF);
        const v8f c = dwt_tile_wmma(p, A);
        if (lane < 16) {
            const int j = j0 + lane;
            dstLo[j] = c[0];                // D[0, lane] = lo
            dstHi[j] = c[1];                // D[1, lane] = hi
        }
    }

    // tail tile (index nFull), owned by one wave
    if ((nFull & 7) == wave) {
        const int j0 = nFull << 4;
        const float* p = src + (2 * (j0 + n) + kx - DWT_HALF);
        const v8f c = dwt_tile_wmma(p, A);
        const int j = j0 + lane;            // lane in [0,16) are valid columns
        if (lane < 16 && j < outLen) {
            dstLo[j] = c[0];
            dstHi[j] = c[1];
        }
    }
}

__global__ __launch_bounds__(256) void dwt_bior35_kernel(
    const float* __restrict__ x, const float* __restrict__ hac,
    float* __restrict__ out, int rows)
{
    __shared__ float s_bufA[BUFA_SZ];   // 65.7 KB
    __shared__ float s_bufB[BUFB_SZ];   // 33.0 KB  (96.5 KB total, < 320 KB/WGP)

    const size_t r = (size_t)blockIdx.x;
    const float* xrow   = x   + r * DWT_N0;
    float*       outLo3 = out + r * DWT_N3;
    float*       outHi1 = out + (size_t)rows *  DWT_N3                    + r * DWT_N1;
    float*       outHi2 = out + (size_t)rows * (DWT_N3 + DWT_N1)          + r * DWT_N2;
    float*       outHi3 = out + (size_t)rows * (DWT_N3 + DWT_N1 + DWT_N2) + r * DWT_N3;

    const int tid  = threadIdx.x;
    const int lane = tid & 31;
    const int wave = tid >> 5;          // 8 wave32's per block

    // ---- async-copy the input row into LDS (ASYNCcnt-tracked, no VGPR staging)
    float* stage = s_bufA + HALO;
    for (int i = tid * 4; i < DWT_N0; i += 256 * 4) {
        const unsigned loff = (unsigned)(uintptr_t)(stage + i);  // LDS byte offset
        asm volatile("global_load_async_to_lds_b128 %0, %1, off"
                     :: "v"(loff), "v"(xrow + i)
                     : "memory");
    }

    // ---- zero the halos / tails while the async copy is in flight ---------
    if (tid < HALO) { s_bufA[tid] = 0.0f; s_bufB[tid] = 0.0f; }
    if (tid < 32)   { s_bufA[HALO + DWT_N0 + tid] = 0.0f; }   // x tail (idx<=16415 read)
    if (tid < 32)   { s_bufB[HALO + DWT_N1 + tid] = 0.0f; }   // lo1 tail (idx<=8223 read)

    // ---- build A matrices (16x4 f32 layout: lanes0-15 K0/K1, lanes16-31 K2/K3)
    const int m    = lane & 15;
    const int koff = (lane >> 4) << 1;
    v2f A[3];
#pragma unroll
    for (int q = 0; q < 3; ++q) {
        const int i0 = 4 * q + koff;
        float a0 = 0.0f, a1 = 0.0f;
        if (m == 0) {                       // row 0: h0 = hac
            a0 = hac[i0];
            a1 = hac[i0 + 1];
        } else if (m == 1) {                // row 1: h1[i] = ((i&1)?+1:-1)*hac[11-i]
            a0 = (( i0      & 1) ? 1.0f : -1.0f) * hac[DWT_L - 1 - i0];
            a1 = (((i0 + 1) & 1) ? 1.0f : -1.0f) * hac[DWT_L - 2 - i0];
        }
        A[q].x = a0;
        A[q].y = a1;
    }

    asm volatile("s_wait_asynccnt 0x0" ::: "memory");
    __syncthreads();

    // Level 1: x (LDS bufA) -> lo1 (LDS bufB), hi1 (global)
    dwt_level_wmma(s_bufA + HALO, DWT_N1, s_bufB + HALO, outHi1, A, lane, wave);
    __syncthreads();

    // Level 2: lo1 (bufB) -> lo2 (bufA, reusing dead x), hi2 (global).
    // Concurrently clear the stale-x tail past lo2 (read up to idx 4127 in L3).
    if (tid < 32) s_bufA[HALO + DWT_N2 + tid] = 0.0f;
    dwt_level_wmma(s_bufB + HALO, DWT_N2, s_bufA + HALO, outHi2, A, lane, wave);
    __syncthreads();

    // Level 3: lo2 (bufA) -> lo3 (global x0), hi3 (global)
    dwt_level_wmma(s_bufA + HALO, DWT_N3, outLo3, outHi3, A, lane, wave);
}

extern "C" void kernel_launch(void* const* d_in, const int* in_sizes, int n_in,
                              void* d_out, int out_size, void* d_ws, size_t ws_size,
                              hipStream_t stream)
{
    const float* x   = (const float*)d_in[0];
    const float* hac = (const float*)d_in[1];
    float*       out = (float*)d_out;
    const int rows = in_sizes[0] / DWT_N0;   // 16 * 64 = 1024 (b,c) rows
    dwt_bior35_kernel<<<rows, 256, 0, stream>>>(x, hac, out, rows);
}